// StemInference_52493090291816
// MI455X (gfx1250) — compile-verified
//
#include <hip/hip_runtime.h>
#include <stdint.h>

#define Bn 8
#define Hh 1024
#define Ww 1024
#define PAD 8
#define HP (Hh + 2*PAD)   // 1040
#define WP (Ww + 2*PAD)   // 1040
#define RADIUS 15.0f
#define THRV 0.4f
#define THRP 1.0f

typedef float        v2f __attribute__((ext_vector_type(2)));
typedef float        v8f __attribute__((ext_vector_type(8)));
typedef unsigned int v4u __attribute__((ext_vector_type(4)));
typedef int          v4i __attribute__((ext_vector_type(4)));
typedef int          v8i __attribute__((ext_vector_type(8)));

// ---------------------------------------------------------------------------
// Tensor Data Mover: 2D tile (tile_w x tile_h elems, f32) global -> LDS.
// D# per CDNA5 ISA (count=1, type=2, data_size=4B,
// no padding / iteration / multicast). Groups 2/3 zero (<=2D tensor).
// clang-23 toolchain: 6-arg builtin (uint32x4, int32x8, int32x4, int32x4,
// int32x8, i32 cpol).
// ---------------------------------------------------------------------------
static __device__ __forceinline__ void tdm_load_2d(uint32_t lds_byte_off,
                                                   const float* gptr,
                                                   int tile_w, int tile_h,
                                                   int row_stride_elems) {
  uint64_t ga = (uint64_t)(uintptr_t)gptr;
  v4u g0;
  g0.x = 1u;                                                 // count=1, user D#
  g0.y = lds_byte_off;                                       // lds_addr
  g0.z = (uint32_t)(ga & 0xFFFFFFFFu);                       // global_addr[31:0]
  g0.w = (uint32_t)((ga >> 32) & 0x1FFFFFFu) | (2u << 30);   // addr[56:32] | type=2

  const uint32_t td0 = 1u << 20, td1 = 1u << 20;             // tensor dims (never OOB)
  const uint64_t s0  = (uint64_t)row_stride_elems;           // dim0 stride (elems)
  v8i g1;
  g1[0] = (int)(2u << 16);                                            // data_size=2 (4B)
  g1[1] = (int)((td0 & 0xFFFFu) << 16);                               // tensor_dim0 lo
  g1[2] = (int)(((td0 >> 16) & 0xFFFFu) | ((td1 & 0xFFFFu) << 16));   // dim0 hi | dim1 lo
  g1[3] = (int)(((td1 >> 16) & 0xFFFFu) | ((uint32_t)tile_w << 16));  // dim1 hi | tile_dim0
  g1[4] = (int)(uint32_t)tile_h;                                      // tile_dim1, tile_dim2=0
  g1[5] = (int)(uint32_t)(s0 & 0xFFFFFFFFu);                          // stride0[31:0]
  g1[6] = (int)(uint32_t)((s0 >> 32) & 0xFFFFu);                      // stride0[47:32], stride1=0
  g1[7] = 0;
  v4i gz4 = {0, 0, 0, 0};
  v8i gz8 = {0, 0, 0, 0, 0, 0, 0, 0};
  __builtin_amdgcn_tensor_load_to_lds(g0, g1, gz4, gz4, gz8, 0);
}

// ---------------------------------------------------------------------------
// K0: zero the workspace (votes + boxed, padded layouts)
// ---------------------------------------------------------------------------
__global__ __launch_bounds__(256) void zero_kernel(float4* ws, long n4) {
  long i = (long)blockIdx.x * 256 + threadIdx.x;
  if (i < n4) ws[i] = make_float4(0.f, 0.f, 0.f, 0.f);
}

// ---------------------------------------------------------------------------
// K1: vote scatter (global_atomic_add_f32 into padded votes)
// ---------------------------------------------------------------------------
__global__ __launch_bounds__(256) void scatter_kernel(const float* __restrict__ kp,
                                                      const float* __restrict__ off,
                                                      float* __restrict__ votes) {
  int p = blockIdx.x * 256 + threadIdx.x;        // 0 .. B*H*W-1
  int b = p / (Hh * Ww);
  int q = p - b * Hh * Ww;
  int y = q / Ww, x = q - y * Ww;
  float w  = kp[p];
  float ox = off[(size_t)(b * 2)     * Hh * Ww + q];
  float oy = off[(size_t)(b * 2 + 1) * Hh * Ww + q];
  int vx = (int)rintf((float)x + RADIUS * ox);   // round-half-even, matches jnp.round
  int vy = (int)rintf((float)y + RADIUS * oy);
  if (vx >= 0 && vx < Ww && vy >= 0 && vy < Hh && w > THRV) {
    atomicAdd(&votes[(size_t)b * HP * WP + (size_t)(vy + PAD) * WP + (vx + PAD)], w);
  }
}

// ---------------------------------------------------------------------------
// K2: fused 9x9 box-sum. TDM stages 40x72 tile+halo into LDS; vertical 9-sum
// into LDS; horizontal 9-sum as banded-ones matmul via V_WMMA_F32_16X16X4_F32.
// Block = 256 thr (8 waves), each wave one 16x16 output tile -> 32x64 region.
// ---------------------------------------------------------------------------
__global__ __launch_bounds__(256) void box_kernel(const float* __restrict__ votes,
                                                  float* __restrict__ boxed) {
  __shared__ float raw[40 * 72];    // rows r0-4..r0+35, cols c0-4..c0+67 (TDM-packed)
  __shared__ float vsum[32 * 73];   // vertical 9-sums, stride 73 (bank-friendly)
  const int b = blockIdx.z, r0 = blockIdx.y * 32, c0 = blockIdx.x * 64;
  const int tid = threadIdx.x;

  if (tid < 32) {  // wave 0 issues the TDM descriptor and waits
    const float* src = votes + (size_t)b * HP * WP
                             + (size_t)(PAD + r0 - 4) * WP + (PAD + c0 - 4);
    tdm_load_2d((uint32_t)(uintptr_t)&raw[0], src, 72, 40, WP);
    __builtin_amdgcn_s_wait_tensorcnt(0);
  }
  __syncthreads();

  // vertical pass: vsum[i][j] = sum_{dy=0..8} raw[i+dy][j]
  for (int idx = tid; idx < 32 * 72; idx += 256) {
    int i = idx / 72, j = idx - i * 72;
    float acc = 0.f;
#pragma unroll
    for (int dy = 0; dy < 9; ++dy) acc += raw[(i + dy) * 72 + j];
    vsum[i * 73 + j] = acc;
  }
  __syncthreads();

  // horizontal pass: out[m][n] = sum_k vsum[m][tc+k] * Band[k][n],
  // Band[k][n] = 1 iff 0 <= k-n <= 8, k in [0,24) -> 6 chained K=4 WMMAs.
  const int wave = tid >> 5, lane = tid & 31;
  const int tr = (wave >> 2) * 16, tc = (wave & 3) * 16;
  const int m = lane & 15, hi = lane >> 4;   // A rows / B,D cols: n == m
  v8f c = {0.f, 0.f, 0.f, 0.f, 0.f, 0.f, 0.f, 0.f};
#pragma unroll
  for (int st = 0; st < 6; ++st) {
    int k0 = 4 * st + 2 * hi;                // lanes 16-31 hold K=2,3 of each step
    v2f a, bb;
    a.x = vsum[(tr + m) * 73 + tc + k0];
    a.y = vsum[(tr + m) * 73 + tc + k0 + 1];
    int d0 = k0 - m;
    bb.x = (d0     >= 0 && d0     <= 8) ? 1.0f : 0.0f;
    bb.y = (d0 + 1 >= 0 && d0 + 1 <= 8) ? 1.0f : 0.0f;
    c = __builtin_amdgcn_wmma_f32_16x16x4_f32(false, a, false, bb,
                                              (short)0, c, false, false);
  }

  // store D: VGPR r -> row m-index r + 8*hi, col = lane&15
  float* dst = boxed + (size_t)b * HP * WP;
#pragma unroll
  for (int r = 0; r < 8; ++r) {
    int row = r0 + tr + r + 8 * hi;
    int col = c0 + tc + m;
    dst[(size_t)(PAD + row) * WP + (PAD + col)] = c[r];
  }
}

// ---------------------------------------------------------------------------
// K3: fused 17x17 max-dilation + peak test. TDM stages 48x80 boxed tile+halo;
// separable max in LDS; isclose(v, dil) && v > 1 -> float mask out.
// ---------------------------------------------------------------------------
__global__ __launch_bounds__(256) void peak_kernel(const float* __restrict__ boxed,
                                                   float* __restrict__ out) {
  __shared__ float raw2[48 * 80];   // rows r0-8..r0+39, cols c0-8..c0+71
  __shared__ float cmax[32 * 81];
  const int b = blockIdx.z, r0 = blockIdx.y * 32, c0 = blockIdx.x * 64;
  const int tid = threadIdx.x;

  if (tid < 32) {
    const float* src = boxed + (size_t)b * HP * WP
                             + (size_t)(PAD + r0 - 8) * WP + (PAD + c0 - 8);
    tdm_load_2d((uint32_t)(uintptr_t)&raw2[0], src, 80, 48, WP);
    __builtin_amdgcn_s_wait_tensorcnt(0);
  }
  __syncthreads();

  for (int idx = tid; idx < 32 * 80; idx += 256) {
    int i = idx / 80, j = idx - i * 80;
    float mx = raw2[i * 80 + j];
#pragma unroll
    for (int dy = 1; dy < 17; ++dy) mx = fmaxf(mx, raw2[(i + dy) * 80 + j]);
    cmax[i * 81 + j] = mx;
  }
  __syncthreads();

#pragma unroll
  for (int t = 0; t < 8; ++t) {
    int o = tid + 256 * t;            // 32x64 outputs per block
    int row = o >> 6, col = o & 63;
    float mx = cmax[row * 81 + col];
#pragma unroll
    for (int dx = 1; dx < 17; ++dx) mx = fmaxf(mx, cmax[row * 81 + col + dx]);
    float v = raw2[(row + 8) * 80 + (col + 8)];
    bool peak = (fabsf(v - mx) <= 1e-8f + 1e-5f * fabsf(mx)) && (v > THRP);
    out[(size_t)b * Hh * Ww + (size_t)(r0 + row) * Ww + (c0 + col)] = peak ? 1.0f : 0.0f;
  }
}

// ---------------------------------------------------------------------------
extern "C" void kernel_launch(void* const* d_in, const int* in_sizes, int n_in,
                              void* d_out, int out_size, void* d_ws, size_t ws_size,
                              hipStream_t stream) {
  const float* kp  = (const float*)d_in[0];   // (B,1,H,W) f32
  const float* off = (const float*)d_in[1];   // (B,2,H,W) f32
  float* votes = (float*)d_ws;                          // B x HP x WP (padded)
  float* boxed = votes + (size_t)Bn * HP * WP;          // B x HP x WP (padded)
  float* outp  = (float*)d_out;                         // (B,H,W) mask as f32

  long n4 = ((long)2 * Bn * HP * WP) / 4;
  zero_kernel<<<(unsigned)((n4 + 255) / 256), 256, 0, stream>>>((float4*)d_ws, n4);

  scatter_kernel<<<(Bn * Hh * Ww) / 256, 256, 0, stream>>>(kp, off, votes);

  dim3 grid(Ww / 64, Hh / 32, Bn);
  box_kernel<<<grid, 256, 0, stream>>>(votes, boxed);
  peak_kernel<<<grid, 256, 0, stream>>>(boxed, outp);
}